// MaxUnpooling2D_54709293416957
// MI455X (gfx1250) — compile-verified
//
#include <hip/hip_runtime.h>

// MaxUnpooling2D inverted to own-the-window gather:
//   each thread owns one pooled cell (b,i,j,c..c+3), reads its update+mask
//   once, and writes all four output positions of its 2x2 SAME window
//   (selected slot gets the value, others get 0). Every output element is
//   written exactly once -> no zero-fill, no atomics, minimum HBM traffic
//   (~617 MB => ~26 us at 23.3 TB/s).
//
// VALU minimization vs previous round:
//  - (b,i,j) come from the launch geometry (no magic-divide chains)
//  - mask decode y=m/(WO*C), x=(m/C)%WO is replaced by the bit-exact
//    equivalent q = m>>8 (== y*224+x, since C==256) compared against
//    q0 = y0*224+x0 + {0,1,224,225}.

typedef float vfloat4 __attribute__((ext_vector_type(4)));
typedef int   vint4   __attribute__((ext_vector_type(4)));

#define B_  8
#define H_  112
#define W_  112
#define C_  256
#define HO_ (2 * H_)
#define WO_ (2 * W_)
#define C4_ (C_ / 4)

static_assert(C_ == 256, "q = mask >> 8 relies on C == 256");
static_assert((W_ % 4) == 0, "grid.x tiles W in groups of 4");

__global__ __launch_bounds__(256)
void maxunpool2d_gather_kernel(const vfloat4* __restrict__ upd,
                               const vint4*   __restrict__ msk,
                               vfloat4*       __restrict__ out)
{
    // block: (c4: 64, jj: 4); grid: (W/4, H, B) -- exact tiling, no guard.
    const int c4 = threadIdx.x;                       // [0,64)
    const int j  = (blockIdx.x << 2) + threadIdx.y;   // [0,112)
    const int i  = blockIdx.y;                        // [0,112)
    const int b  = blockIdx.z;                        // [0,8)

    // Input flat index in float4/int4 units: ((b*H + i)*W + j)*C4 + c4.
    const int tid = (((b * H_ + i) * W_ + j) << 6) + c4;

    const vfloat4 u = __builtin_nontemporal_load(&upd[tid]);
    const vint4   m = __builtin_nontemporal_load(&msk[tid]);

    const int y0 = 2 * i;
    const int x0 = 2 * j;
    const int q0 = y0 * WO_ + x0;   // target row-major pixel id of window TL

    vfloat4 r00, r01, r10, r11;
#pragma unroll
    for (int k = 0; k < 4; ++k) {
        // Bit-exact reference decode: q = (y*WO + x) for any mask value,
        // channel bits of the mask intentionally ignored (lane supplies c).
        const int d = (int)((unsigned)m[k] >> 8) - q0;
        const float v = u[k];
        r00[k] = (d == 0)        ? v : 0.0f;
        r01[k] = (d == 1)        ? v : 0.0f;
        r10[k] = (d == WO_)      ? v : 0.0f;
        r11[k] = (d == WO_ + 1)  ? v : 0.0f;
    }

    // Output base in float4 units at (b, y0, x0, c4); window neighbors are
    // immediate-offset reachable (row pitch 224*64*16B = 229376 B < 2^23).
    const size_t rowp = (size_t)WO_ * C4_;
    const size_t base = (((size_t)b * HO_ + (size_t)y0) * WO_ + (size_t)x0) * C4_
                      + (size_t)c4;

    __builtin_nontemporal_store(r00, &out[base]);
    __builtin_nontemporal_store(r01, &out[base + C4_]);
    __builtin_nontemporal_store(r10, &out[base + rowp]);
    __builtin_nontemporal_store(r11, &out[base + rowp + C4_]);
}

extern "C" void kernel_launch(void* const* d_in, const int* in_sizes, int n_in,
                              void* d_out, int out_size, void* d_ws, size_t ws_size,
                              hipStream_t stream)
{
    const vfloat4* upd = (const vfloat4*)d_in[0];  // updates f32 [B,H,W,C]
    const vint4*   msk = (const vint4*)d_in[1];    // mask    i32 [B,H,W,C]
    vfloat4*       out = (vfloat4*)d_out;          // f32 [B,2H,2W,C]

    dim3 block(C4_, 4, 1);           // 256 threads = 8 wave32
    dim3 grid(W_ / 4, H_, B_);       // exact cover of (b,i,j)

    maxunpool2d_gather_kernel<<<grid, block, 0, stream>>>(upd, msk, out);
}